// JSNN_87041807221062
// MI455X (gfx1250) — compile-verified
//
#include <hip/hip_runtime.h>

#define N_NODES   4000
#define N_LAYERS  8
#define PER_LAYER 500
#define IN_DEG    32
#define BATCH     2048
#define IN_DIM    784
#define OUT_DIM   10
#define EDGES_PER_LAYER (PER_LAYER * IN_DEG)   // 16000

typedef __attribute__((ext_vector_type(2))) float v2f;
typedef __attribute__((ext_vector_type(8))) float v8f;

// Async global->LDS copy (gfx1250): VDST = wave-relative LDS byte offset,
// VADDR = 64-bit global address, SADDR = off. Tracked by ASYNCcnt.
__device__ __forceinline__ void async_ld_b32(unsigned lds_off, const float* g)
{
    asm volatile("global_load_async_to_lds_b32 %0, %1, off"
                 :: "v"(lds_off), "v"(g)
                 : "memory");
}

// ---------------------------------------------------------------------------
// Kernel 1: h[v,b] = relu(sum_k W_in[v,k]*x[b,k] + b_in[v])  via f32 WMMA.
// Block = 256 threads = 8 waves (2x4), block tile 32(M=vertex) x 64(N=batch).
// K in chunks of 16, DOUBLE-BUFFERED via global_load_async_to_lds_b32:
// stage chunk c+1 into buffer p^1 while WMMAs consume buffer p.
// ---------------------------------------------------------------------------
#define KC 16
#define NCH (IN_DIM / KC)   // 49

__global__ __launch_bounds__(256)
void in_gemm_wmma(const float* __restrict__ x, const float* __restrict__ W,
                  const float* __restrict__ bias, float* __restrict__ h)
{
    __shared__ float As[2][32][KC + 1];   // W tile   [m][k], padded
    __shared__ float Bs[2][KC][64 + 1];   // x^T tile [k][n], padded
    const int tid  = threadIdx.x;
    const int lane = tid & 31;
    const int wave = tid >> 5;
    const int wm = wave >> 2;             // 0..1
    const int wn = wave & 3;              // 0..3
    const int m0 = blockIdx.y * 32;
    const int n0 = blockIdx.x * 64;
    const int hi = lane >> 4;             // 0/1 : half-wave
    const int lo = lane & 15;

    // Wave-relative LDS byte offsets of the staging buffers (generic shared
    // pointers truncate to the LDS offset per the flat->LDS aperture rules).
    const unsigned ldsA = (unsigned)(uintptr_t)&As[0][0][0];
    const unsigned ldsB = (unsigned)(uintptr_t)&Bs[0][0][0];

    // Issue one K-chunk stage: 2 A-elements + 4 B-elements per thread,
    // 6 async-load instructions per wave. Coalesced along k (A) / k (B).
    auto stage = [&](int k0, int p) {
        #pragma unroll
        for (int r = 0; r < 2; ++r) {
            int idx = tid + r * 256;
            int row = idx >> 4, col = idx & 15;
            int m = m0 + row;
            m = (m < PER_LAYER) ? m : (PER_LAYER - 1);   // clamp: rows >=500 unused
            async_ld_b32(ldsA + (unsigned)(((p * 32 + row) * (KC + 1) + col) * 4),
                         W + m * IN_DIM + (k0 + col));
        }
        #pragma unroll
        for (int r = 0; r < 4; ++r) {
            int idx = tid + r * 256;
            int n = idx >> 4, j = idx & 15;
            async_ld_b32(ldsB + (unsigned)(((p * KC + j) * (64 + 1) + n) * 4),
                         x + (n0 + n) * IN_DIM + (k0 + j));
        }
    };

    v8f acc = {};

    stage(0, 0);                               // prologue: fill buffer 0
    for (int c = 0; c < NCH; ++c) {
        const int p = c & 1;
        if (c + 1 < NCH) {
            stage((c + 1) * KC, p ^ 1);        // overlap: fill other buffer
            // 6 outstanding = the stage just issued; buffer p is complete.
            asm volatile("s_wait_asynccnt 0x6" ::: "memory");
        } else {
            asm volatile("s_wait_asynccnt 0x0" ::: "memory");
        }
        __syncthreads();                       // buffer p visible to all waves

        #pragma unroll
        for (int j4 = 0; j4 < KC / 4; ++j4) {
            const int kb = j4 * 4 + 2 * hi;    // K = r + 2*(lane/16) per ISA layout
            const int m  = wm * 16 + lo;
            const int n  = wn * 16 + lo;
            v2f a, b;
            a.x = As[p][m][kb];   a.y = As[p][m][kb + 1];
            b.x = Bs[p][kb][n];   b.y = Bs[p][kb + 1][n];
            acc = __builtin_amdgcn_wmma_f32_16x16x4_f32(
                false, a, false, b, (short)0, acc, false, false);
        }
        __syncthreads();                       // all waves done with buffer p
    }

    // epilogue: bias + relu, C layout: row = r + 8*hi, col = lane%16
    const int col = n0 + wn * 16 + lo;
    #pragma unroll
    for (int r = 0; r < 8; ++r) {
        int m = m0 + wm * 16 + r + 8 * hi;
        if (m < PER_LAYER) {
            float v = acc[r] + bias[m];
            h[(long)m * BATCH + col] = v > 0.0f ? v : 0.0f;
        }
    }
}

// ---------------------------------------------------------------------------
// Kernel 2 (one launch per layer): block = one target vertex, 256 threads
// cover all 2048 batch columns (2 x float4 each). 32 (src,w) pairs in LDS.
// Gathered rows are L2-resident (h = 32MB < 192MB L2); prefetch next row.
// ---------------------------------------------------------------------------
__global__ __launch_bounds__(256)
void layer_agg(float* __restrict__ h, const int* __restrict__ edge_src,
               const float* __restrict__ w_edge, int layer)
{
    __shared__ int   s_src[IN_DEG];
    __shared__ float s_w[IN_DEG];
    const int v   = blockIdx.x;
    const int tid = threadIdx.x;
    const int e0  = (layer - 1) * EDGES_PER_LAYER + v * IN_DEG;
    if (tid < IN_DEG) {
        s_src[tid] = edge_src[e0 + tid];
        s_w[tid]   = w_edge[e0 + tid];
    }
    __syncthreads();

    const int c0 = tid * 4;                       // cols [c0,c0+4) and [1024+c0,...)
    float4 acc0 = {0.f, 0.f, 0.f, 0.f};
    float4 acc1 = {0.f, 0.f, 0.f, 0.f};

    #pragma unroll 4
    for (int e = 0; e < IN_DEG; ++e) {
        const float w = s_w[e];
        const float* row = h + (long)s_src[e] * BATCH;
        if (e + 1 < IN_DEG)
            __builtin_prefetch(h + (long)s_src[e + 1] * BATCH + c0, 0, 0);
        float4 u = *(const float4*)(row + c0);
        float4 t = *(const float4*)(row + 1024 + c0);
        acc0.x = fmaf(w, u.x, acc0.x); acc0.y = fmaf(w, u.y, acc0.y);
        acc0.z = fmaf(w, u.z, acc0.z); acc0.w = fmaf(w, u.w, acc0.w);
        acc1.x = fmaf(w, t.x, acc1.x); acc1.y = fmaf(w, t.y, acc1.y);
        acc1.z = fmaf(w, t.z, acc1.z); acc1.w = fmaf(w, t.w, acc1.w);
    }
    acc0.x = acc0.x > 0.f ? acc0.x : 0.f;  acc0.y = acc0.y > 0.f ? acc0.y : 0.f;
    acc0.z = acc0.z > 0.f ? acc0.z : 0.f;  acc0.w = acc0.w > 0.f ? acc0.w : 0.f;
    acc1.x = acc1.x > 0.f ? acc1.x : 0.f;  acc1.y = acc1.y > 0.f ? acc1.y : 0.f;
    acc1.z = acc1.z > 0.f ? acc1.z : 0.f;  acc1.w = acc1.w > 0.f ? acc1.w : 0.f;

    float* dst = h + (long)(layer * PER_LAYER + v) * BATCH;
    *(float4*)(dst + c0)        = acc0;
    *(float4*)(dst + 1024 + c0) = acc1;
}

// ---------------------------------------------------------------------------
// Kernel 3: out[b,o] = sum_v W_out[o,v]*h[out_verts[v],b] + b_out[o].
// Thread = one batch column; out_verts/W_out chunk-staged in LDS.
// ---------------------------------------------------------------------------
#define CHUNK 128

__global__ __launch_bounds__(256)
void out_gemm(const float* __restrict__ h, const float* __restrict__ Wout,
              const float* __restrict__ bout, const int* __restrict__ out_verts,
              int v_out, float* __restrict__ out)
{
    __shared__ int   s_idx[CHUNK];
    __shared__ float s_w[OUT_DIM][CHUNK];
    const int tid = threadIdx.x;
    const int b   = blockIdx.x * 256 + tid;

    float acc[OUT_DIM];
    #pragma unroll
    for (int o = 0; o < OUT_DIM; ++o) acc[o] = bout[o];

    for (int v0 = 0; v0 < v_out; v0 += CHUNK) {
        const int nv = (v_out - v0) < CHUNK ? (v_out - v0) : CHUNK;
        __syncthreads();
        if (tid < CHUNK && tid < nv) s_idx[tid] = out_verts[v0 + tid];
        for (int idx = tid; idx < OUT_DIM * CHUNK; idx += 256) {
            int o = idx >> 7, vv = idx & (CHUNK - 1);
            if (vv < nv) s_w[o][vv] = Wout[o * v_out + v0 + vv];
        }
        __syncthreads();
        for (int vv = 0; vv < nv; ++vv) {
            float hv = h[(long)s_idx[vv] * BATCH + b];   // coalesced across lanes
            #pragma unroll
            for (int o = 0; o < OUT_DIM; ++o)
                acc[o] = fmaf(s_w[o][vv], hv, acc[o]);
        }
    }
    #pragma unroll
    for (int o = 0; o < OUT_DIM; ++o) out[b * OUT_DIM + o] = acc[o];
}

// ---------------------------------------------------------------------------
extern "C" void kernel_launch(void* const* d_in, const int* in_sizes, int n_in,
                              void* d_out, int out_size, void* d_ws, size_t ws_size,
                              hipStream_t stream)
{
    const float* x        = (const float*)d_in[0];
    const float* W_in     = (const float*)d_in[1];
    const float* b_in     = (const float*)d_in[2];
    const float* w_edge   = (const float*)d_in[3];
    const float* W_out    = (const float*)d_in[4];
    const float* b_out    = (const float*)d_in[5];
    const int*   edge_src = (const int*)d_in[6];
    // d_in[7] = edge_dst_local, d_in[8] = edge_offsets: structure is static
    // (layer-l edges are the contiguous 32 at (l-1)*16000 + v*32), so unused.
    const int*   out_verts = (const int*)d_in[9];
    const int    v_out     = in_sizes[9];

    float* h   = (float*)d_ws;                 // [N_NODES][BATCH] fp32 = 32 MB
    float* out = (float*)d_out;                // [BATCH][OUT_DIM]

    // 1) input GEMM + relu -> h[0:500]   (512 padded rows / 32 = 16 M-tiles)
    in_gemm_wmma<<<dim3(BATCH / 64, 16), 256, 0, stream>>>(x, W_in, b_in, h);

    // 2) 7 dependent sparse aggregation layers
    for (int l = 1; l < N_LAYERS; ++l)
        layer_agg<<<PER_LAYER, 256, 0, stream>>>(h, edge_src, w_edge, l);

    // 3) readout GEMM
    out_gemm<<<BATCH / 256, 256, 0, stream>>>(h, W_out, b_out, out_verts, v_out, out);
}